// PCLLosses_5025111736796
// MI455X (gfx1250) — compile-verified
//
#include <hip/hip_runtime.h>

#define KCLUST 1024
#define BLOCK  256
#define EPS_F      1e-9f
#define CLAMP_HI_F 1e4f

__device__ __forceinline__ float clipf(float x) {
    return fminf(fmaxf(x, EPS_F), CLAMP_HI_F);
}

__global__ void pcl_init_ws(float* __restrict__ ws, int n) {
    int i = blockIdx.x * blockDim.x + threadIdx.x;
    if (i < n) ws[i] = 0.0f;
}

__global__ __launch_bounds__(BLOCK) void pcl_main(
    const float* __restrict__ probs,
    const int*   __restrict__ labels,
    const float* __restrict__ wts,
    const int*   __restrict__ gts,
    float* __restrict__ g_sum,
    float* __restrict__ g_cnt,
    float* __restrict__ g_sumw,
    float* __restrict__ g_bg,
    long long n, int c)
{
    __shared__ float lds_sum[KCLUST];
    __shared__ float lds_cnt[KCLUST];
    __shared__ float lds_sumw[KCLUST];
    __shared__ float lds_stage[2][BLOCK];   // double-buffered async staging

    const int tid = threadIdx.x;
    for (int k = tid; k < KCLUST; k += BLOCK) {
        lds_sum[k]  = 0.0f;
        lds_cnt[k]  = 0.0f;
        lds_sumw[k] = 0.0f;
    }
    __syncthreads();

    const long long stride = (long long)gridDim.x * BLOCK;
    const long long i0 = (long long)blockIdx.x * BLOCK + tid;

    // LDS byte offsets of this lane's two staging slots (flat-ptr low 32 bits).
    const unsigned slot0 = (unsigned)(unsigned long long)&lds_stage[0][tid];
    const unsigned slot1 = (unsigned)(unsigned long long)&lds_stage[1][tid];

    // Prologue: kick off the first weight transfer on the ASYNCcnt path.
    if (i0 < n) {
        unsigned long long ga = (unsigned long long)(wts + i0);
        asm volatile("global_load_async_to_lds_b32 %0, %1, off"
                     :: "v"(slot0), "v"(ga) : "memory");
    }

    float bg_acc = 0.0f;
    int buf = 0;
    for (long long i = i0; i < n; i += stride) {
        const long long inext = i + stride;
        const bool has_next = inext < n;

        // Software pipeline: issue next tile's weight into the other buffer
        // before touching this tile, so the DMA overlaps the scattered gather.
        if (has_next) {
            unsigned slot = buf ? slot0 : slot1;
            unsigned long long ga = (unsigned long long)(wts + inext);
            asm volatile("global_load_async_to_lds_b32 %0, %1, off"
                         :: "v"(slot), "v"(ga) : "memory");
        }

        int gt  = __builtin_nontemporal_load(gts + i);
        int lab = __builtin_nontemporal_load(labels + i);
        long long col = (gt < 0) ? 0 : (long long)lab;
        float v = clipf(__builtin_nontemporal_load(probs + i * (long long)c + col));

        // Async loads complete in order: <=1 outstanding proves the current
        // buffer landed while the next transfer stays in flight.
        if (has_next) asm volatile("s_wait_asynccnt 0x1" ::: "memory");
        else          asm volatile("s_wait_asynccnt 0x0" ::: "memory");
        float w = lds_stage[buf][tid];
        buf ^= 1;

        if (gt < 0) {
            bg_acc += logf(v) * w;             // negated at finalize
        } else {
            atomicAdd(&lds_sum[gt],  v);       // ds_add_f32 (no return)
            atomicAdd(&lds_cnt[gt],  1.0f);
            atomicAdd(&lds_sumw[gt], w);
        }
    }

    // wave32 butterfly reduction of the background partial
    for (int off = 16; off > 0; off >>= 1)
        bg_acc += __shfl_xor(bg_acc, off, 32);
    if ((tid & 31) == 0)
        atomicAdd(g_bg, bg_acc);

    __syncthreads();
    // Flush per-block segment sums; only touched clusters hit global atomics.
    for (int k = tid; k < KCLUST; k += BLOCK) {
        float cg = lds_cnt[k];
        if (cg != 0.0f) {
            atomicAdd(&g_sum[k],  lds_sum[k]);
            atomicAdd(&g_cnt[k],  cg);
            atomicAdd(&g_sumw[k], lds_sumw[k]);
        }
    }
}

__global__ __launch_bounds__(KCLUST) void pcl_finalize(
    const float* __restrict__ g_sum,
    const float* __restrict__ g_cnt,
    const float* __restrict__ g_sumw,
    const float* __restrict__ g_bg,
    float* __restrict__ out, long long n)
{
    __shared__ float partial[KCLUST / 32];
    const int tid = threadIdx.x;

    float cg = g_cnt[tid];
    float contrib = 0.0f;
    if (cg > 0.0f)
        contrib = logf(g_sum[tid] / cg) * g_sumw[tid];   // empty clusters -> 0

    for (int off = 16; off > 0; off >>= 1)
        contrib += __shfl_xor(contrib, off, 32);
    if ((tid & 31) == 0) partial[tid >> 5] = contrib;
    __syncthreads();

    if (tid < 32) {
        float v = partial[tid];
        for (int off = 16; off > 0; off >>= 1)
            v += __shfl_xor(v, off, 32);
        if (tid == 0)
            out[0] = -(v + g_bg[0]) / (float)n;
    }
}

extern "C" void kernel_launch(void* const* d_in, const int* in_sizes, int n_in,
                              void* d_out, int out_size, void* d_ws, size_t ws_size,
                              hipStream_t stream) {
    const float* probs  = (const float*)d_in[0];   // [N, C] f32
    const int*   labels = (const int*)  d_in[1];   // [N] i32
    const float* wts    = (const float*)d_in[2];   // [N] f32
    const int*   gts    = (const int*)  d_in[3];   // [N] i32

    long long n = (long long)in_sizes[1];
    int c = (int)(in_sizes[0] / (n > 0 ? n : 1));

    float* ws     = (float*)d_ws;
    float* g_sum  = ws;
    float* g_cnt  = ws + KCLUST;
    float* g_sumw = ws + 2 * KCLUST;
    float* g_bg   = ws + 3 * KCLUST;

    const int init_n = 3 * KCLUST + 1;
    pcl_init_ws<<<(init_n + 255) / 256, 256, 0, stream>>>(ws, init_n);

    const int blocks = 1024;
    pcl_main<<<blocks, BLOCK, 0, stream>>>(probs, labels, wts, gts,
                                           g_sum, g_cnt, g_sumw, g_bg, n, c);

    pcl_finalize<<<1, KCLUST, 0, stream>>>(g_sum, g_cnt, g_sumw, g_bg,
                                           (float*)d_out, n);
}